// Decoder_38147899523225
// MI455X (gfx1250) — compile-verified
//
#include <hip/hip_runtime.h>
#include <hip/hip_bf16.h>
#include <math.h>

// Problem dims
#define BB 64
#define PP 196
#define EE 2048
#define DD 512
#define VV 20000
#define LL 26
#define TT 25

#if defined(__gfx1250__) && __has_builtin(__builtin_amdgcn_wmma_f32_16x16x4_f32)
#define USE_WMMA_F32 1
typedef __attribute__((ext_vector_type(2))) float v2f;
typedef __attribute__((ext_vector_type(8))) float v8f;
#endif

// ---------------------------------------------------------------------------
// Stable argsort by decreasing caption length; gather caps; emit dec_len.
// Single block of 64 threads.
// ---------------------------------------------------------------------------
__global__ void k_sort(const int* __restrict__ cap_len,
                       int* __restrict__ order,
                       int* __restrict__ declen,
                       const int* __restrict__ caps_in,
                       int* __restrict__ caps_s,
                       float* __restrict__ out_caps,
                       float* __restrict__ out_dlen) {
  int b = threadIdx.x;
  if (b < BB) {
    int my = cap_len[b];
    int rank = 0;
    for (int j = 0; j < BB; ++j) {
      int lj = cap_len[j];
      if (lj > my || (lj == my && j < b)) rank++;
    }
    order[rank] = b;
  }
  __syncthreads();
  if (b < BB) {
    int src = order[b];
    int dl = cap_len[src] - 1;
    declen[b] = dl;
    out_dlen[b] = (float)dl;
    for (int t = 0; t < LL; ++t) {
      int cv = caps_in[src * LL + t];
      caps_s[b * LL + t] = cv;
      out_caps[b * LL + t] = (float)cv;
    }
  }
}

// ---------------------------------------------------------------------------
// avg[b][e] = mean_p enc[order[b]][p][e]
// ---------------------------------------------------------------------------
__global__ void k_avg(const float* __restrict__ enc,
                      const int* __restrict__ order,
                      float* __restrict__ avg) {
  int idx = blockIdx.x * blockDim.x + threadIdx.x;
  if (idx >= BB * EE) return;
  int b = idx / EE, e = idx % EE;
  const float* src = enc + (long long)order[b] * PP * EE + e;
  float s = 0.f;
  for (int p = 0; p < PP; ++p) s += src[(long long)p * EE];
  avg[idx] = s * (1.0f / PP);
}

// ---------------------------------------------------------------------------
// fp32 WMMA GEMM: one wave per 32 x (16*NT) tile (MT=2 row-tiles reuse every
// B fragment). NT (1..4) and TB (B transposed) are compile-time ->
// branch-free K-loop; all loads of a k-step are grouped before the 2*NT
// wmmas so they pipeline instead of serializing wait->wmma.
//   - optional row indirection on A: src_row = order[r/rpb]*rpb + r%rpb
//   - TB: B element (k,n) read as Bm[n*ldb + k]   (for W.T GEMMs)
//   - optional mask epilogue: C = (mask[row] > mt) ? val : 0
// Grid: blockIdx.x = mtile32 * Nc + nchunk; columns start at tn_base.
// A frag (16x4 fp32, 2 VGPR): lane<16 -> M=lane, k=kk+{0,1};
//                             lane>=16 -> M=lane-16, k=kk+{2,3}.
// B frag (4x16 fp32, 2 VGPR): mirrored (rows kk+{0,1} / kk+{2,3}).
// C/D (8 VGPR): VGPR r -> M = tmBase + r + 8*(lane>=16), N = tn + (lane&15).
// ---------------------------------------------------------------------------
template <int NT, bool TB>
__global__ __launch_bounds__(32)
void k_gemm(const float* __restrict__ A, int lda,
            const int* __restrict__ rord, int rpb,
            const float* __restrict__ Bm, int ldb,
            const float* __restrict__ bias,
            float* __restrict__ C, long long ldc,
            int K, int Nc, int tn_base,
            const int* __restrict__ mask, int mt) {
  int tile = blockIdx.x;
  int tm = (tile / Nc) << 5;                    // 32 rows per wave
  int tn0 = tn_base + (tile % Nc) * (NT << 4);
  int lane = threadIdx.x & 31;
  int half = lane >> 4;   // 0 or 1
  int l15 = lane & 15;

#if USE_WMMA_F32
  int koff = half << 1;   // 0 or 2
  int ar0 = tm + l15;
  int ar1 = tm + 16 + l15;
  long long arow0 = rord ? ((long long)rord[ar0 / rpb] * rpb + (ar0 % rpb))
                         : (long long)ar0;
  long long arow1 = rord ? ((long long)rord[ar1 / rpb] * rpb + (ar1 % rpb))
                         : (long long)ar1;
  const float* Ap0 = A + arow0 * (long long)lda;
  const float* Ap1 = A + arow1 * (long long)lda;

  v8f acc0[NT], acc1[NT];
  #pragma unroll
  for (int nt = 0; nt < NT; ++nt)
    #pragma unroll
    for (int r = 0; r < 8; ++r) {
      acc0[nt][r] = 0.0f;
      acc1[nt][r] = 0.0f;
    }

  #pragma unroll 2
  for (int kk = 0; kk < K; kk += 4) {
    // ---- load phase: 2 A fragments + NT B fragments ----
    v2f a0, a1, bv[NT];
    a0.x = Ap0[kk + koff];
    a0.y = Ap0[kk + koff + 1];
    a1.x = Ap1[kk + koff];
    a1.y = Ap1[kk + koff + 1];
    #pragma unroll
    for (int nt = 0; nt < NT; ++nt) {
      int col = tn0 + (nt << 4) + l15;
      if (TB) {
        const float* Bp = Bm + (long long)col * ldb + kk + koff;
        bv[nt].x = Bp[0];
        bv[nt].y = Bp[1];
      } else {
        const float* Bp = Bm + (long long)(kk + koff) * ldb + col;
        bv[nt].x = Bp[0];
        bv[nt].y = Bp[ldb];
      }
    }
    // ---- math phase: each B fragment feeds two wmmas (A-row reuse) ----
    #pragma unroll
    for (int nt = 0; nt < NT; ++nt) {
      acc0[nt] = __builtin_amdgcn_wmma_f32_16x16x4_f32(
          false, a0, false, bv[nt], (short)0, acc0[nt], false, false);
      acc1[nt] = __builtin_amdgcn_wmma_f32_16x16x4_f32(
          false, a1, false, bv[nt], (short)0, acc1[nt], false, false);
    }
  }

  #pragma unroll
  for (int nt = 0; nt < NT; ++nt) {
    int col = tn0 + (nt << 4) + l15;
    float bs = bias ? bias[col] : 0.f;
    #pragma unroll
    for (int r = 0; r < 8; ++r) {
      int row0 = tm + r + (half << 3);
      int row1 = row0 + 16;
      float v0 = acc0[nt][r] + bs;
      float v1 = acc1[nt][r] + bs;
      if (mask) {
        v0 = (mask[row0] > mt) ? v0 : 0.f;
        v1 = (mask[row1] > mt) ? v1 : 0.f;
      }
      C[(long long)row0 * ldc + col] = v0;
      C[(long long)row1 * ldc + col] = v1;
    }
  }
#else
  // Scalar fallback (host pass / builtin missing): same tiling, plain FMA.
  for (int mtile = 0; mtile < 2; ++mtile) {
    for (int nt = 0; nt < NT; ++nt) {
      int col = tn0 + (nt << 4) + l15;
      float bs = bias ? bias[col] : 0.f;
      for (int r = 0; r < 8; ++r) {
        int row = tm + (mtile << 4) + r + (half << 3);
        long long arow = rord
            ? ((long long)rord[row / rpb] * rpb + (row % rpb))
            : (long long)row;
        const float* Ap = A + arow * (long long)lda;
        float s = 0.f;
        for (int k = 0; k < K; ++k) {
          float bvv = TB ? Bm[(long long)col * ldb + k]
                         : Bm[(long long)k * ldb + col];
          s += Ap[k] * bvv;
        }
        float v = s + bs;
        if (mask) v = (mask[row] > mt) ? v : 0.f;
        C[(long long)row * ldc + col] = v;
      }
    }
  }
#endif
}

// ---------------------------------------------------------------------------
// att[b][p] = b_att + sum_d relu(enc_proj[b][p][d] + dec_p[b][d]) * w_att[d]
// One wave per (b,p); 8 waves per block.
// ---------------------------------------------------------------------------
__global__ __launch_bounds__(256)
void k_att(const float* __restrict__ enc_proj,
           const float* __restrict__ dec_p,
           const float* __restrict__ w_att,
           const float* __restrict__ b_att,
           float* __restrict__ att) {
  int wid = blockIdx.x * 8 + (threadIdx.x >> 5);
  if (wid >= BB * PP) return;
  int lane = threadIdx.x & 31;
  int b = wid / PP;
  const float* ep = enc_proj + (long long)wid * DD;
  const float* dp = dec_p + (long long)b * DD;
  float s = 0.f;
  for (int d = lane; d < DD; d += 32) {
    float v = ep[d] + dp[d];
    v = fmaxf(v, 0.f);
    s += v * w_att[d];
  }
  for (int off = 16; off > 0; off >>= 1) s += __shfl_down(s, off, 32);
  if (lane == 0) att[wid] = s + b_att[0];
}

// ---------------------------------------------------------------------------
// Per batch row: softmax(att) -> alpha (write masked to out), context
// awe[e] = sum_p alpha[p]*enc[order[b]][p][e], gated by sigmoid(gate_pre),
// and assemble x = [emb[caps[b][t]], gate*awe].
// One block (256 threads) per b.
// ---------------------------------------------------------------------------
__global__ __launch_bounds__(256)
void k_ctx(const float* __restrict__ att,
           const float* __restrict__ enc,
           const int* __restrict__ order,
           const float* __restrict__ gate_pre,
           const float* __restrict__ embt,
           const int* __restrict__ caps_s,
           const int* __restrict__ declen, int t,
           float* __restrict__ x,
           float* __restrict__ out_alpha) {
  __shared__ float salpha[PP];
  __shared__ float red[256];
  int b = blockIdx.x;
  int tid = threadIdx.x;
  const float* ab = att + b * PP;

  float m = -1e30f;
  for (int p = tid; p < PP; p += 256) m = fmaxf(m, ab[p]);
  red[tid] = m;
  __syncthreads();
  for (int s = 128; s > 0; s >>= 1) {
    if (tid < s) red[tid] = fmaxf(red[tid], red[tid + s]);
    __syncthreads();
  }
  m = red[0];
  __syncthreads();

  float sum = 0.f;
  for (int p = tid; p < PP; p += 256) {
    float e = expf(ab[p] - m);
    salpha[p] = e;
    sum += e;
  }
  red[tid] = sum;
  __syncthreads();
  for (int s = 128; s > 0; s >>= 1) {
    if (tid < s) red[tid] += red[tid + s];
    __syncthreads();
  }
  float inv = 1.f / red[0];
  __syncthreads();

  int active = declen[b] > t;
  for (int p = tid; p < PP; p += 256) {
    salpha[p] *= inv;
    out_alpha[((long long)b * TT + t) * PP + p] = active ? salpha[p] : 0.f;
  }
  __syncthreads();

  // x[0:D] = embedding of token t
  const float* et = embt + (long long)caps_s[b * LL + t] * DD;
  float* xb = x + (long long)b * (DD + EE);
  for (int d = tid; d < DD; d += 256) xb[d] = et[d];

  // x[D:D+E] = sigmoid(gate_pre) * (alpha . enc)
  const float* eb = enc + (long long)order[b] * PP * EE;
  const float* gb = gate_pre + (long long)b * EE;
  for (int e = tid; e < EE; e += 256) {
    float s2 = 0.f;
    for (int p = 0; p < PP; ++p) s2 += salpha[p] * eb[(long long)p * EE + e];
    float g = 1.f / (1.f + expf(-gb[e]));
    xb[DD + e] = g * s2;
  }
}

// ---------------------------------------------------------------------------
// LSTM pointwise: gates g = g1 + g2 (chunks i,f,g,o of size D).
// Update h,c only where active (dec_len sorted desc -> prefix-active).
// ---------------------------------------------------------------------------
__global__ __launch_bounds__(256)
void k_lstm(const float* __restrict__ g1,
            const float* __restrict__ g2,
            const int* __restrict__ declen, int t,
            float* __restrict__ h, float* __restrict__ c) {
  int idx = blockIdx.x * blockDim.x + threadIdx.x;
  if (idx >= BB * DD) return;
  int b = idx / DD, d = idx % DD;
  const float* G1 = g1 + (long long)b * 4 * DD;
  const float* G2 = g2 + (long long)b * 4 * DD;
  float ig = G1[d] + G2[d];
  float fg = G1[DD + d] + G2[DD + d];
  float gg = G1[2 * DD + d] + G2[2 * DD + d];
  float og = G1[3 * DD + d] + G2[3 * DD + d];
  float si = 1.f / (1.f + expf(-ig));
  float sf = 1.f / (1.f + expf(-fg));
  float so = 1.f / (1.f + expf(-og));
  float tg = tanhf(gg);
  float cn = sf * c[idx] + si * tg;
  float hn = so * tanhf(cn);
  if (declen[b] > t) {
    c[idx] = cn;
    h[idx] = hn;
  }
}

// ---------------------------------------------------------------------------
// Host-side GEMM dispatcher: full 64-wide chunks with NT=4, remainder chunk
// (16/32/48 cols) with NT=1/2/3. transB picked at compile time. M must be a
// multiple of 32 (holds for all call sites: 64 and 12544).
// ---------------------------------------------------------------------------
struct GemmArgs {
  const float* A; int lda;
  const int* rord; int rpb;
  const float* Bm; int ldb;
  const float* bias;
  float* C; long long ldc;
  int K;
  const int* mask; int mt;
};

template <int NT>
static void launch_nt(bool tb, int grid, int Nc, int tn_base,
                      const GemmArgs& g, hipStream_t stream) {
  if (tb)
    k_gemm<NT, true><<<dim3(grid), dim3(32), 0, stream>>>(
        g.A, g.lda, g.rord, g.rpb, g.Bm, g.ldb, g.bias, g.C, g.ldc, g.K, Nc,
        tn_base, g.mask, g.mt);
  else
    k_gemm<NT, false><<<dim3(grid), dim3(32), 0, stream>>>(
        g.A, g.lda, g.rord, g.rpb, g.Bm, g.ldb, g.bias, g.C, g.ldc, g.K, Nc,
        tn_base, g.mask, g.mt);
}

static void run_gemm(const GemmArgs& g, bool tb, int M, int N,
                     hipStream_t stream) {
  int Mt = M / 32;                 // 32-row wave tiles
  int full = N / 64;
  int rem = (N - full * 64) / 16;  // 0..3 sub-tiles of 16
  if (full > 0) launch_nt<4>(tb, Mt * full, full, 0, g, stream);
  if (rem == 1) launch_nt<1>(tb, Mt, 1, full * 64, g, stream);
  else if (rem == 2) launch_nt<2>(tb, Mt, 1, full * 64, g, stream);
  else if (rem == 3) launch_nt<3>(tb, Mt, 1, full * 64, g, stream);
}

// ---------------------------------------------------------------------------
extern "C" void kernel_launch(void* const* d_in, const int* in_sizes, int n_in,
                              void* d_out, int out_size, void* d_ws, size_t ws_size,
                              hipStream_t stream) {
  const float* enc   = (const float*)d_in[0];
  const int*   caps  = (const int*)d_in[1];
  const int*   clen  = (const int*)d_in[2];
  const float* W_enc = (const float*)d_in[3];
  const float* b_enc = (const float*)d_in[4];
  const float* W_dec = (const float*)d_in[5];
  const float* b_dec = (const float*)d_in[6];
  const float* w_att = (const float*)d_in[7];
  const float* b_att = (const float*)d_in[8];
  const float* W_h   = (const float*)d_in[9];
  const float* b_h   = (const float*)d_in[10];
  const float* W_c   = (const float*)d_in[11];
  const float* b_c   = (const float*)d_in[12];
  const float* W_fb  = (const float*)d_in[13];
  const float* b_fb  = (const float*)d_in[14];
  const float* W_ih  = (const float*)d_in[15];  // [4D, D+E] row-major
  const float* b_ih  = (const float*)d_in[16];
  const float* W_hh  = (const float*)d_in[17];  // [4D, D] row-major
  const float* b_hh  = (const float*)d_in[18];
  const float* W_fc  = (const float*)d_in[19];  // [D, V]
  const float* b_fc  = (const float*)d_in[20];
  const float* embt  = (const float*)d_in[21];  // [V, D]

  float* out = (float*)d_out;
  float* out_pred  = out;                                   // [B,T,V]
  float* out_caps  = out + (long long)BB * TT * VV;         // [B,L]
  float* out_dlen  = out_caps + BB * LL;                    // [B]
  float* out_alpha = out_dlen + BB;                         // [B,T,P]

  char* ws = (char*)d_ws;
  size_t off = 0;
  auto alloc = [&](size_t bytes) -> void* {
    void* p = ws + off;
    off += (bytes + 255) & ~(size_t)255;
    return p;
  };
  int*   order  = (int*)alloc(BB * sizeof(int));
  int*   declen = (int*)alloc(BB * sizeof(int));
  int*   caps_s = (int*)alloc((size_t)BB * LL * sizeof(int));
  float* avg    = (float*)alloc((size_t)BB * EE * 4);
  float* h      = (float*)alloc((size_t)BB * DD * 4);
  float* c      = (float*)alloc((size_t)BB * DD * 4);
  float* encp   = (float*)alloc((size_t)BB * PP * DD * 4);
  float* decp   = (float*)alloc((size_t)BB * DD * 4);
  float* att    = (float*)alloc((size_t)BB * PP * 4);
  float* gatep  = (float*)alloc((size_t)BB * EE * 4);
  float* x      = (float*)alloc((size_t)BB * (DD + EE) * 4);
  float* g1     = (float*)alloc((size_t)BB * 4 * DD * 4);
  float* g2     = (float*)alloc((size_t)BB * 4 * DD * 4);
  (void)ws_size; (void)in_sizes; (void)n_in; (void)out_size;

  // ---- prologue ----
  k_sort<<<dim3(1), dim3(64), 0, stream>>>(clen, order, declen, caps, caps_s,
                                           out_caps, out_dlen);
  k_avg<<<dim3((BB * EE + 255) / 256), dim3(256), 0, stream>>>(enc, order, avg);
  {
    GemmArgs g{avg, EE, nullptr, 1, W_h, DD, b_h, h, DD, EE, nullptr, 0};
    run_gemm(g, false, BB, DD, stream);
  }
  {
    GemmArgs g{avg, EE, nullptr, 1, W_c, DD, b_c, c, DD, EE, nullptr, 0};
    run_gemm(g, false, BB, DD, stream);
  }
  {  // enc_proj over sorted rows via indirection: src row = order[r/P]*P + r%P
    GemmArgs g{enc, EE, order, PP, W_enc, DD, b_enc, encp, DD, EE, nullptr, 0};
    run_gemm(g, false, BB * PP, DD, stream);
  }

  // ---- decode loop ----
  for (int t = 0; t < TT; ++t) {
    {
      GemmArgs g{h, DD, nullptr, 1, W_dec, DD, b_dec, decp, DD, DD,
                 nullptr, 0};
      run_gemm(g, false, BB, DD, stream);
    }
    k_att<<<dim3((BB * PP + 7) / 8), dim3(256), 0, stream>>>(encp, decp, w_att,
                                                             b_att, att);
    {
      GemmArgs g{h, DD, nullptr, 1, W_fb, EE, b_fb, gatep, EE, DD,
                 nullptr, 0};
      run_gemm(g, false, BB, EE, stream);
    }
    k_ctx<<<dim3(BB), dim3(256), 0, stream>>>(att, enc, order, gatep, embt,
                                              caps_s, declen, t, x, out_alpha);
    {  // g1 = x @ W_ih.T + b_ih   (TB: B(k,n) = W_ih[n*(D+E)+k])
      GemmArgs g{x, DD + EE, nullptr, 1, W_ih, DD + EE, b_ih, g1, 4 * DD,
                 DD + EE, nullptr, 0};
      run_gemm(g, true, BB, 4 * DD, stream);
    }
    {  // g2 = h @ W_hh.T + b_hh
      GemmArgs g{h, DD, nullptr, 1, W_hh, DD, b_hh, g2, 4 * DD, DD,
                 nullptr, 0};
      run_gemm(g, true, BB, 4 * DD, stream);
    }
    k_lstm<<<dim3((BB * DD + 255) / 256), dim3(256), 0, stream>>>(g1, g2,
                                                                  declen, t,
                                                                  h, c);
    {  // preds = h_new @ W_fc + b_fc, masked to 0 where inactive, written
       // into predictions[:, t, :] (row stride T*V). N=20000 -> 312 full
       // 64-wide chunks (NT=4) + one 32-wide remainder (NT=2).
      GemmArgs g{h, DD, nullptr, 1, W_fc, VV, b_fc,
                 out_pred + (long long)t * VV, (long long)TT * VV, DD,
                 declen, t};
      run_gemm(g, false, BB, VV, stream);
    }
  }
}